// CriticEstimator_77764677862009
// MI455X (gfx1250) — compile-verified
//
#include <hip/hip_runtime.h>
#include <hip/hip_bf16.h>
#include <math.h>

// Problem constants (from reference)
#define BB 8192
#define NN 16
#define SS 128
#define AA 64
#define EE 256
#define HH 8
#define DD 32
#define KIN 192   // S + A
#define K2 512    // 2*E

typedef __attribute__((ext_vector_type(2))) float v2f;
typedef __attribute__((ext_vector_type(8))) float v8f;

__device__ __forceinline__ v8f wmma_f32(v2f a, v2f b, v8f c) {
    // V_WMMA_F32_16X16X4_F32 : D = A(16x4) x B(4x16) + C(16x16)
    return __builtin_amdgcn_wmma_f32_16x16x4_f32(
        /*neg_a=*/false, a, /*neg_b=*/false, b,
        /*c_mod=*/(short)0, c, /*reuse_a=*/false, /*reuse_b=*/false);
}

// ---------------------------------------------------------------------------
// Kernel 1: e[b,n,:] = concat(state,action)[b,n,:] @ g_weight[n] + g_bias[n]
// grid (B/64, N), block 128 (4 waves). Wave w owns output cols [64w, 64w+64).
// 4 row-tiles (64 batch rows): each weight fragment feeds 4 WMMAs.
// __launch_bounds__(128,1): allow ~200 VGPRs so the 128 accumulator VGPRs
// do NOT spill to scratch (spills observed with the default budget).
// ---------------------------------------------------------------------------
__global__ __launch_bounds__(128, 1)
void ce_gemm_e(const float* __restrict__ state,
               const float* __restrict__ action,
               const float* __restrict__ g_w,
               const float* __restrict__ g_b,
               float* __restrict__ e_buf) {
    __shared__ float xin[64][196];   // 192 + pad (stride 196: 16 rows -> 16 banks)
    const int b0 = blockIdx.x * 64;
    const int n  = blockIdx.y;
    const int tid = threadIdx.x;

    for (int i = tid; i < 64 * 128; i += 128) {
        int r = i >> 7, k = i & 127;
        xin[r][k] = state[((size_t)(b0 + r) * NN + n) * SS + k];
    }
    for (int i = tid; i < 64 * 64; i += 128) {
        int r = i >> 6, k = i & 63;
        xin[r][128 + k] = action[((size_t)(b0 + r) * NN + n) * AA + k];
    }
    __syncthreads();

    const int wave = tid >> 5;
    const int lane = tid & 31;
    const int row  = lane & 15;
    const int hi   = lane >> 4;
    const int ko   = hi * 2;
    const int colBase = wave * 64;

    v8f acc[4][4] = {};              // [rowtile][coltile]
    for (int kc = 0; kc < KIN; kc += 32) {
        {   // unconditional chunk-ahead prefetch (speculative; OOB dropped)
            const float* pf =
                g_w + ((size_t)n * KIN + (kc + 32 + ko)) * EE + colBase + (lane & 15);
            __builtin_prefetch(pf, 0, 1);
            __builtin_prefetch(pf + 32, 0, 1);
        }
#pragma unroll
        for (int k0 = 0; k0 < 32; k0 += 4) {
            const int k = kc + k0;
            v2f a[4];
#pragma unroll
            for (int rt = 0; rt < 4; ++rt) {
                a[rt].x = xin[rt * 16 + row][k + ko];
                a[rt].y = xin[rt * 16 + row][k + ko + 1];
            }
#pragma unroll
            for (int t = 0; t < 4; ++t) {
                int col = colBase + t * 16 + (lane & 15);
                const float* wp = g_w + ((size_t)n * KIN + (k + ko)) * EE + col;
                v2f b; b.x = wp[0]; b.y = wp[EE];
#pragma unroll
                for (int rt = 0; rt < 4; ++rt)
                    acc[rt][t] = wmma_f32(a[rt], b, acc[rt][t]);
            }
        }
    }
#pragma unroll
    for (int t = 0; t < 4; ++t) {
        int col = colBase + t * 16 + (lane & 15);
        float bias = g_b[n * EE + col];
#pragma unroll
        for (int rt = 0; rt < 4; ++rt)
#pragma unroll
            for (int r = 0; r < 8; ++r) {
                int m = rt * 16 + r + 8 * hi;
                e_buf[((size_t)(b0 + m) * NN + n) * EE + col] = acc[rt][t][r] + bias;
            }
    }
}

// ---------------------------------------------------------------------------
// Kernel 2: fused QKV + attention row-sum, x = t * relu(V)
// grid (B/2), block 256 (8 waves). Wave h = head h; 2 batches per block share
// every W_Q/W_K/W_V fragment (2 row-tiles). Dynamic LDS ~103 KB (<320 KB/WGP).
// ---------------------------------------------------------------------------
#define SMEM2_FLOATS (2 * 16 * 258 + 8 * 2 * 2 * 16 * 34)
__global__ __launch_bounds__(256, 1)
void ce_attn(const float* __restrict__ e_buf,
             const float* __restrict__ W_Q,
             const float* __restrict__ W_K,
             const float* __restrict__ W_V,
             float* __restrict__ x_buf) {
    extern __shared__ float smem[];
    float (*e_s)[16][258] = (float(*)[16][258])smem;                 // [g][n][c]
    float (*qk)[2][2][16][34] = (float(*)[2][2][16][34])(smem + 2 * 16 * 258); // [h][QK][g][n][d]
    const int b0  = blockIdx.x * 2;
    const int tid = threadIdx.x;

    for (int i = tid; i < 2 * 16 * 256; i += 256) {
        int g = i >> 12, r = (i >> 8) & 15, c = i & 255;
        e_s[g][r][c] = e_buf[((size_t)(b0 + g) * NN + r) * EE + c];
    }
    __syncthreads();

    const int h    = tid >> 5;
    const int lane = tid & 31;
    const int row  = lane & 15;
    const int hi   = lane >> 4;
    const int ko   = hi * 2;

    // ---- Q and K projections, 2 batches x 2 column tiles ----
    v8f q[2][2] = {}, kk[2][2] = {};
    for (int k0 = 0; k0 < EE; k0 += 4) {
        v2f a[2];
#pragma unroll
        for (int g = 0; g < 2; ++g) {
            a[g].x = e_s[g][row][k0 + ko];
            a[g].y = e_s[g][row][k0 + ko + 1];
        }
#pragma unroll
        for (int t = 0; t < 2; ++t) {
            int col = h * DD + t * 16 + (lane & 15);
            const v2f bq = *(const v2f*)(W_Q + (size_t)col * EE + k0 + ko);
            const v2f bk = *(const v2f*)(W_K + (size_t)col * EE + k0 + ko);
#pragma unroll
            for (int g = 0; g < 2; ++g) {
                q[g][t]  = wmma_f32(a[g], bq, q[g][t]);
                kk[g][t] = wmma_f32(a[g], bk, kk[g][t]);
            }
        }
    }
#pragma unroll
    for (int g = 0; g < 2; ++g)
#pragma unroll
        for (int t = 0; t < 2; ++t)
#pragma unroll
            for (int r = 0; r < 8; ++r) {
                int m  = r + 8 * hi;
                int cl = t * 16 + (lane & 15);
                qk[h][0][g][m][cl] = q[g][t][r];
                qk[h][1][g][m][cl] = kk[g][t][r];
            }
    __syncthreads();

    // ---- attn = Q K^T / sqrt(D); softmax row-sum (faithful) ----
    const float scale = 0.1767766952966369f;   // 1/sqrt(32)
    float t_r[2][8];
#pragma unroll
    for (int g = 0; g < 2; ++g) {
        v8f att = {};
        for (int d0 = 0; d0 < DD; d0 += 4) {
            v2f a, bv;
            a.x  = qk[h][0][g][row][d0 + ko];
            a.y  = qk[h][0][g][row][d0 + ko + 1];
            bv.x = qk[h][1][g][lane & 15][d0 + ko];     // B[k=d, col=m] = K[m,d]
            bv.y = qk[h][1][g][lane & 15][d0 + ko + 1];
            att = wmma_f32(a, bv, att);
        }
#pragma unroll
        for (int r = 0; r < 8; ++r) {
            float v = att[r] * scale;
            float mx = v;
#pragma unroll
            for (int m = 1; m <= 8; m <<= 1) mx = fmaxf(mx, __shfl_xor(mx, m, 32));
            float ex = __expf(v - mx);
            float Z = ex;
#pragma unroll
            for (int m = 1; m <= 8; m <<= 1) Z += __shfl_xor(Z, m, 32);
            float p = ex / Z;
            float t = p;
#pragma unroll
            for (int m = 1; m <= 8; m <<= 1) t += __shfl_xor(t, m, 32);
            t_r[g][r] = t;
        }
    }

    // ---- V projection, x = t * relu(V) ----
    v8f vv[2][2] = {};
    for (int k0 = 0; k0 < EE; k0 += 4) {
        v2f a[2];
#pragma unroll
        for (int g = 0; g < 2; ++g) {
            a[g].x = e_s[g][row][k0 + ko];
            a[g].y = e_s[g][row][k0 + ko + 1];
        }
#pragma unroll
        for (int t = 0; t < 2; ++t) {
            int col = h * DD + t * 16 + (lane & 15);
            const v2f bv = *(const v2f*)(W_V + (size_t)col * EE + k0 + ko);
#pragma unroll
            for (int g = 0; g < 2; ++g)
                vv[g][t] = wmma_f32(a[g], bv, vv[g][t]);
        }
    }
#pragma unroll
    for (int g = 0; g < 2; ++g)
#pragma unroll
        for (int t = 0; t < 2; ++t)
#pragma unroll
            for (int r = 0; r < 8; ++r) {
                int m   = r + 8 * hi;
                int col = h * DD + t * 16 + (lane & 15);
                float xv = fmaxf(vv[g][t][r], 0.0f) * t_r[g][r];
                x_buf[((size_t)(b0 + g) * NN + m) * EE + col] = xv;
            }
}

// ---------------------------------------------------------------------------
// Kernel 3: out[b,n] = gx_w2[n] . relu([e|x] @ gx_w1[n] + b1) + b2[n]
// grid (B/64, N), block 256 (8 waves). Wave w owns fc1 cols [32w, 32w+32),
// 4 row-tiles per block. Dynamic LDS ~129 KB (<320 KB/WGP, 2 blocks/WGP).
// ---------------------------------------------------------------------------
#define SMEM3_FLOATS (64 * 516 + 64)
__global__ __launch_bounds__(256, 1)
void ce_fc(const float* __restrict__ e_buf,
           const float* __restrict__ x_buf,
           const float* __restrict__ gx_w1,
           const float* __restrict__ gx_b1,
           const float* __restrict__ gx_w2,
           const float* __restrict__ gx_b2,
           float* __restrict__ out) {
    extern __shared__ float smem[];
    float (*ex)[516] = (float(*)[516])smem;      // [row][k], k in [0,512)
    float* outAcc = smem + 64 * 516;
    const int b0  = blockIdx.x * 64;
    const int n   = blockIdx.y;
    const int tid = threadIdx.x;

    for (int i = tid; i < 64 * 256; i += 256) {
        int r = i >> 8, c = i & 255;
        size_t base = ((size_t)(b0 + r) * NN + n) * EE + c;
        ex[r][c]       = e_buf[base];
        ex[r][256 + c] = x_buf[base];
    }
    if (tid < 64) outAcc[tid] = 0.0f;
    __syncthreads();

    const int w    = tid >> 5;
    const int lane = tid & 31;
    const int row  = lane & 15;
    const int hi   = lane >> 4;
    const int ko   = hi * 2;

    v8f f[4][2] = {};                // [rowtile][coltile]
    for (int kc = 0; kc < K2; kc += 32) {
        {   // unconditional chunk-ahead prefetch (speculative; OOB dropped)
            const float* pf =
                gx_w1 + ((size_t)n * K2 + (kc + 32 + ko)) * EE + w * 32 + (lane & 15);
            __builtin_prefetch(pf, 0, 1);
            __builtin_prefetch(pf + 16, 0, 1);
        }
#pragma unroll
        for (int k0 = 0; k0 < 32; k0 += 4) {
            const int k = kc + k0;
            v2f a[4];
#pragma unroll
            for (int rt = 0; rt < 4; ++rt) {
                a[rt].x = ex[rt * 16 + row][k + ko];
                a[rt].y = ex[rt * 16 + row][k + ko + 1];
            }
#pragma unroll
            for (int t = 0; t < 2; ++t) {
                int col = w * 32 + t * 16 + (lane & 15);
                const float* wp = gx_w1 + ((size_t)n * K2 + (k + ko)) * EE + col;
                v2f b; b.x = wp[0]; b.y = wp[EE];
#pragma unroll
                for (int rt = 0; rt < 4; ++rt)
                    f[rt][t] = wmma_f32(a[rt], b, f[rt][t]);
            }
        }
    }

    // fc2: per-row dot over this wave's 32 columns, LDS atomic combine
    float b1[2], w2[2];
#pragma unroll
    for (int t = 0; t < 2; ++t) {
        int col = w * 32 + t * 16 + (lane & 15);
        b1[t] = gx_b1[n * EE + col];
        w2[t] = gx_w2[n * EE + col];     // gx_weight_2 is [N, E, 1]
    }
#pragma unroll
    for (int rt = 0; rt < 4; ++rt)
#pragma unroll
        for (int r = 0; r < 8; ++r) {
            float partial = 0.0f;
#pragma unroll
            for (int t = 0; t < 2; ++t) {
                float fr = fmaxf(f[rt][t][r] + b1[t], 0.0f);
                partial += fr * w2[t];
            }
#pragma unroll
            for (int m = 1; m <= 8; m <<= 1) partial += __shfl_xor(partial, m, 32);
            if ((lane & 15) == 0)
                atomicAdd(&outAcc[rt * 16 + r + 8 * hi], partial);  // ds_add_f32
        }
    __syncthreads();
    if (tid < 64)
        out[(size_t)(b0 + tid) * NN + n] = outAcc[tid] + gx_b2[n];
}

// ---------------------------------------------------------------------------
extern "C" void kernel_launch(void* const* d_in, const int* in_sizes, int n_in,
                              void* d_out, int out_size, void* d_ws, size_t ws_size,
                              hipStream_t stream) {
    const float* state  = (const float*)d_in[0];
    const float* action = (const float*)d_in[1];
    const float* g_w    = (const float*)d_in[2];
    const float* g_b    = (const float*)d_in[3];
    const float* W_Q    = (const float*)d_in[4];
    const float* W_K    = (const float*)d_in[5];
    const float* W_V    = (const float*)d_in[6];
    const float* gx_w1  = (const float*)d_in[7];
    const float* gx_b1  = (const float*)d_in[8];
    const float* gx_w2  = (const float*)d_in[9];
    const float* gx_b2  = (const float*)d_in[10];
    float* out = (float*)d_out;

    // workspace: e [B,N,E] then x [B,N,E]  (2 * 128 MiB)
    float* e_buf = (float*)d_ws;
    float* x_buf = e_buf + (size_t)BB * NN * EE;

    dim3 g1(BB / 64, NN, 1);
    ce_gemm_e<<<g1, 128, 0, stream>>>(state, action, g_w, g_b, e_buf);

    ce_attn<<<BB / 2, 256, SMEM2_FLOATS * sizeof(float), stream>>>(
        e_buf, W_Q, W_K, W_V, x_buf);

    dim3 g3(BB / 64, NN, 1);
    ce_fc<<<g3, 256, SMEM3_FLOATS * sizeof(float), stream>>>(
        e_buf, x_buf, gx_w1, gx_b1, gx_w2, gx_b2, out);
}